// _NormalizerModule_84361747628501
// MI455X (gfx1250) — compile-verified
//
#include <hip/hip_runtime.h>
#include <stdint.h>

// ---------------------------------------------------------------------------
// Segmented mean-subtract (memory-bound, ~360MB traffic -> ~16us @ 23.3TB/s).
// CDNA5 paths: GLOBAL_LOAD_ASYNC_TO_LDS_B128 (ASYNCcnt) tile staging,
// s_wait_asynccnt, wave32 LDS run-length scan of sorted segment ids,
// boundary-only global_atomic_add_f32 (L2-resident 3.2MB accumulator),
// non-temporal hints on the use-once streams to keep the mean table in L2.
// ---------------------------------------------------------------------------

#define BLK   256           // 8 waves (wave32) per workgroup
#define CHUNK 8             // contiguous atoms scanned per thread
#define TILE  (BLK * CHUNK) // 2048 atoms per block tile

// gfx1250 flat aperture: generic pointer into LDS carries the LDS byte offset
// in addr[31:0] (ISA 10.2 aperture table) -> truncate to get the DS address.
__device__ __forceinline__ uint32_t lds_addr_of(const void* p) {
  return (uint32_t)(uintptr_t)p;
}

__device__ __forceinline__ void async_b128_to_lds(uint32_t lds, const void* gaddr) {
  unsigned long long ga = (unsigned long long)(uintptr_t)gaddr;
  asm volatile("global_load_async_to_lds_b128 %0, %1, off"
               :: "v"(lds), "v"(ga)
               : "memory");
}

__device__ __forceinline__ void wait_async_all() {
  asm volatile("s_wait_asynccnt 0" ::: "memory");
}

__device__ __forceinline__ void atomic_add_agent(float* p, float v) {
  // result unused -> lowers to non-returning global_atomic_add_f32 (STOREcnt)
  __hip_atomic_fetch_add(p, v, __ATOMIC_RELAXED, __HIP_MEMORY_SCOPE_AGENT);
}

// ---------------------------------------------------------------------------
// Kernel 1: zero the per-molecule accumulator {sx, sy, sz, count} (4 floats).
// num_mols lives in device memory (scalar input) -> device-side bound.
// ---------------------------------------------------------------------------
__global__ void zero_ws(float* __restrict__ w, const int* __restrict__ num_mols_p) {
  const int total = (*num_mols_p) * 4;
  for (int i = blockIdx.x * blockDim.x + threadIdx.x; i < total;
       i += gridDim.x * blockDim.x)
    __builtin_nontemporal_store(0.0f, &w[i]);
}

// ---------------------------------------------------------------------------
// Kernel 2: segment accumulate. Full tiles: async global->LDS staging, then
// per-thread run-length scan of 8 contiguous atoms; atomics only on id change.
// ---------------------------------------------------------------------------
__global__ void __launch_bounds__(BLK)
seg_accum(const float* __restrict__ x, const int* __restrict__ gb,
          float* __restrict__ w, int N) {
  __shared__ __align__(16) float s_xyz[TILE * 3]; // 24 KB
  __shared__ __align__(16) int   s_id[TILE];      //  8 KB

  const int tid = threadIdx.x;
  const long long base = (long long)blockIdx.x * TILE;

  if (base + TILE <= (long long)N) {
    // ---- stage coords: TILE*3 floats = 1536 x b128, 6 per thread ----
    const float* gx = x + base * 3;
    for (int i = tid; i < (TILE * 3) / 4; i += BLK)
      async_b128_to_lds(lds_addr_of(&s_xyz[i * 4]), gx + i * 4);
    // ---- stage ids: TILE ints = 512 x b128, 2 per thread ----
    const int* gi = gb + base;
    for (int i = tid; i < TILE / 4; i += BLK)
      async_b128_to_lds(lds_addr_of(&s_id[i * 4]), gi + i * 4);

    wait_async_all();     // this wave's ASYNCcnt -> 0
    __syncthreads();      // all waves staged

    // ---- run-length accumulate over this thread's contiguous chunk ----
    const int j0 = tid * CHUNK;
    int   cur = s_id[j0];
    float ax = 0.f, ay = 0.f, az = 0.f, ac = 0.f;
    #pragma unroll
    for (int k = 0; k < CHUNK; ++k) {
      const int id = s_id[j0 + k];
      if (id != cur) {
        float* d = w + (size_t)cur * 4;
        atomic_add_agent(d + 0, ax);
        atomic_add_agent(d + 1, ay);
        atomic_add_agent(d + 2, az);
        atomic_add_agent(d + 3, ac);
        cur = id; ax = ay = az = ac = 0.f;
      }
      ax += s_xyz[(j0 + k) * 3 + 0];
      ay += s_xyz[(j0 + k) * 3 + 1];
      az += s_xyz[(j0 + k) * 3 + 2];
      ac += 1.0f;
    }
    float* d = w + (size_t)cur * 4;
    atomic_add_agent(d + 0, ax);
    atomic_add_agent(d + 1, ay);
    atomic_add_agent(d + 2, az);
    atomic_add_agent(d + 3, ac);
  } else {
    // tail tile: direct atomics, bounds-checked
    for (long long j = base + tid; j < (long long)N; j += BLK) {
      const int id = gb[j];
      float* d = w + (size_t)id * 4;
      atomic_add_agent(d + 0, x[j * 3 + 0]);
      atomic_add_agent(d + 1, x[j * 3 + 1]);
      atomic_add_agent(d + 2, x[j * 3 + 2]);
      atomic_add_agent(d + 3, 1.0f);
    }
  }
}

// ---------------------------------------------------------------------------
// Kernel 3: sums -> means in place. mean = sum / max(count, 1) (empty -> 0).
// Regular-temporal access: keeps the 3.2MB table warm in L2 for seg_apply.
// ---------------------------------------------------------------------------
__global__ void seg_mean(float* __restrict__ w, const int* __restrict__ num_mols_p) {
  const int m = *num_mols_p;
  float4* v4 = (float4*)w;
  for (int i = blockIdx.x * blockDim.x + threadIdx.x; i < m;
       i += gridDim.x * blockDim.x) {
    float4 t = v4[i];
    const float inv = 1.0f / fmaxf(t.w, 1.0f);
    t.x *= inv; t.y *= inv; t.z *= inv; t.w = 0.0f;
    v4[i] = t;
  }
}

// ---------------------------------------------------------------------------
// Kernel 4: out[i] = x[i] - mean[gb[i]]. x/out are use-once streams -> NT
// loads/stores (TH=NT) so they don't evict the hot mean table; mean fetched
// as one global_load_b128.
// ---------------------------------------------------------------------------
__global__ void seg_apply(const float* __restrict__ x, const int* __restrict__ gb,
                          const float* __restrict__ w, float* __restrict__ out,
                          int N) {
  const float4* w4 = (const float4*)w;
  for (int i = blockIdx.x * blockDim.x + threadIdx.x; i < N;
       i += gridDim.x * blockDim.x) {
    const int id = gb[i];
    const float4 m = w4[id];                       // RT: stays in L2
    const float x0 = __builtin_nontemporal_load(&x[i * 3 + 0]);
    const float x1 = __builtin_nontemporal_load(&x[i * 3 + 1]);
    const float x2 = __builtin_nontemporal_load(&x[i * 3 + 2]);
    __builtin_nontemporal_store(x0 - m.x, &out[i * 3 + 0]);
    __builtin_nontemporal_store(x1 - m.y, &out[i * 3 + 1]);
    __builtin_nontemporal_store(x2 - m.z, &out[i * 3 + 2]);
  }
}

// ---------------------------------------------------------------------------
extern "C" void kernel_launch(void* const* d_in, const int* in_sizes, int n_in,
                              void* d_out, int out_size, void* d_ws, size_t ws_size,
                              hipStream_t stream) {
  const float* atoms_x   = (const float*)d_in[0];   // [N,3] f32
  const int*   graph_b   = (const int*)d_in[1];     // [N] int
  const int*   num_mols  = (const int*)d_in[2];     // scalar (device)
  float*       out       = (float*)d_out;           // [N,3] f32
  float*       w         = (float*)d_ws;            // [num_mols,4] accumulators

  const int N = in_sizes[0] / 3;

  // 1) zero accumulators (ws is poisoned; must re-init every deterministic call)
  zero_ws<<<1024, BLK, 0, stream>>>(w, num_mols);

  // 2) segment sums + counts (async-LDS tiles, run-length boundary atomics)
  const int nTiles = (N + TILE - 1) / TILE;
  seg_accum<<<nTiles, BLK, 0, stream>>>(atoms_x, graph_b, w, N);

  // 3) sums -> means
  seg_mean<<<1024, BLK, 0, stream>>>(w, num_mols);

  // 4) subtract (streaming NT pass)
  const int nApply = (N + BLK - 1) / BLK;
  seg_apply<<<nApply, BLK, 0, stream>>>(atoms_x, graph_b, w, out, N);
}